// SPDUnVectorize_38199439131089
// MI455X (gfx1250) — compile-verified
//
#include <hip/hip_runtime.h>
#include <hip/hip_bf16.h>
#include <stdint.h>

// SPDUnVectorize: out[b,i,j] = x[b, packed(min(i,j), max(i,j))]
// n = 512, D = 131328, B = 256.  Pure data movement, ~403 MB -> ~17us HBM bound.
//
// v2: amortize index math + vectorize.
//  - 32x32 tiles over the 16x16 tile grid (136 pairs), 8 batches per block.
//  - off-diagonal: 32 async B32 copies global->LDS per thread-block-batch set
//    (CDNA5 global_load_async_to_lds_b32; offset: applies to BOTH global and
//    LDS address per CDNA5 ISA, so one base serves 4 words),
//    single s_wait_asynccnt + barrier, then float4 NT stores for both the
//    upper block (ds_read_b128) and mirrored lower block (4 strided LDS reads).
//  - diagonal tiles: direct mirrored gather + float4 NT store.

#define N_DIM    512
#define D_PACK   131328          // 512*513/2
#define NT2      16              // tiles per dim (32-wide tiles)
#define NPAIR2   136             // NT2*(NT2+1)/2
#define NB       8               // batches per block
#define ROW_W    36              // LDS row stride in words (16B-aligned, skewed)
#define TILE_WDS (32 * ROW_W)    // 1152 words per batch buffer

typedef float v4f __attribute__((ext_vector_type(4)));

__device__ __forceinline__ int tri_base(int r) {
    return r * N_DIM - (r * (r - 1)) / 2;   // start of packed row r
}

__global__ void __launch_bounds__(256)
spd_unvec_kernel(const float* __restrict__ x, float* __restrict__ out) {
    __shared__ __align__(16) float tile[NB * TILE_WDS];   // 36 KB

    const int tid = threadIdx.x;
    const int r   = tid >> 3;        // tile row   0..31
    const int q   = tid & 7;         // quad index 0..7 (cols 4q..4q+3)
    const int b0  = blockIdx.y * NB; // first batch of this block

    // decode pair index -> (ti, tj), ti <= tj (uniform scalar loop, <=16 iters)
    int rem = blockIdx.x;
    int ti = 0;
    while (rem >= NT2 - ti) { rem -= NT2 - ti; ++ti; }
    const int tj = ti + rem;

    if (ti != tj) {
        // ---------------- off-diagonal tile pair ----------------
        const int i   = ti * 32 + r;             // global row of upper block
        const int j0  = tj * 32 + q * 4;         // first global col of quad
        const int pk  = tri_base(i) + (j0 - i);  // packed offset of quad start

        // issue all NB*4 async B32 copies (ASYNCcnt <= 32 per wave)
#pragma unroll
        for (int bb = 0; bb < NB; ++bb) {
            const float* gp = x + ((size_t)(b0 + bb) * D_PACK + pk);
            unsigned lofs = (unsigned)(uintptr_t)(const void*)
                &tile[bb * TILE_WDS + r * ROW_W + q * 4];
            asm volatile("global_load_async_to_lds_b32 %0, %1, off"
                         :: "v"(lofs), "v"(gp) : "memory");
            asm volatile("global_load_async_to_lds_b32 %0, %1, off offset:4"
                         :: "v"(lofs), "v"(gp) : "memory");
            asm volatile("global_load_async_to_lds_b32 %0, %1, off offset:8"
                         :: "v"(lofs), "v"(gp) : "memory");
            asm volatile("global_load_async_to_lds_b32 %0, %1, off offset:12"
                         :: "v"(lofs), "v"(gp) : "memory");
        }
        asm volatile("s_wait_asynccnt 0x0" ::: "memory");
        __syncthreads();

#pragma unroll
        for (int bb = 0; bb < NB; ++bb) {
            const size_t ob  = (size_t)(b0 + bb) * N_DIM * N_DIM;
            const float* buf = &tile[bb * TILE_WDS];

            // upper block (ti,tj): row-major float4, coalesced NT store
            v4f v = *(const v4f*)&buf[r * ROW_W + q * 4];
            __builtin_nontemporal_store(
                v, (v4f*)&out[ob + (size_t)i * N_DIM + j0]);

            // lower block (tj,ti): transposed gather from LDS, NT store
            // out[tj*32+r][ti*32+4q+k] = tile[4q+k][r]
            v4f tv;
            tv.x = buf[(q * 4 + 0) * ROW_W + r];
            tv.y = buf[(q * 4 + 1) * ROW_W + r];
            tv.z = buf[(q * 4 + 2) * ROW_W + r];
            tv.w = buf[(q * 4 + 3) * ROW_W + r];
            __builtin_nontemporal_store(
                tv, (v4f*)&out[ob + (size_t)(tj * 32 + r) * N_DIM + ti * 32 + q * 4]);
        }
    } else {
        // ---------------- diagonal tile ----------------
        const int gi = ti * 32 + r;              // output row
        const int gj0 = ti * 32 + q * 4;         // first output col of quad
#pragma unroll
        for (int bb = 0; bb < NB; ++bb) {
            const size_t xb = (size_t)(b0 + bb) * D_PACK;
            const size_t ob = (size_t)(b0 + bb) * N_DIM * N_DIM;
            v4f v;
#pragma unroll
            for (int k = 0; k < 4; ++k) {
                const int gj = gj0 + k;
                const int rr = gi < gj ? gi : gj;
                const int cc = gi < gj ? gj : gi;
                float e = x[xb + (size_t)(tri_base(rr) + (cc - rr))];
                if (k == 0) v.x = e; else if (k == 1) v.y = e;
                else if (k == 2) v.z = e; else v.w = e;
            }
            __builtin_nontemporal_store(
                v, (v4f*)&out[ob + (size_t)gi * N_DIM + gj0]);
        }
    }
}

extern "C" void kernel_launch(void* const* d_in, const int* in_sizes, int n_in,
                              void* d_out, int out_size, void* d_ws, size_t ws_size,
                              hipStream_t stream) {
    (void)in_sizes; (void)n_in; (void)out_size; (void)d_ws; (void)ws_size;
    const float* x = (const float*)d_in[0];
    float* out = (float*)d_out;

    dim3 grid(NPAIR2, 256 / NB, 1);   // 136 tile pairs x 32 batch groups
    dim3 block(256, 1, 1);            // 8 waves
    spd_unvec_kernel<<<grid, block, 0, stream>>>(x, out);
}